// KANLayer_52561809768771
// MI455X (gfx1250) — compile-verified
//
#include <hip/hip_runtime.h>
#include <cstdint>

#ifndef __has_builtin
#define __has_builtin(x) 0
#endif

namespace {

constexpr int kI = 64;                                     // inner features
constexpr int kN = 16;                                     // basis fns per feature
constexpr int kO = 64;                                     // output features
constexpr int kBlock = 256;                                // 8 wave32s
constexpr int kTileRows = 16;                              // (b,s) rows per block
constexpr int kRowsPerThread = kTileRows / (kBlock / kO);  // 4
constexpr int kCI = 4;                                     // i-values per staged chunk
constexpr int kChunkFloats = kCI * kN * kO;                // 4096 floats = 16 KB
constexpr int kNChunks = kI / kCI;                         // 16
constexpr int kCopiesPerArray = kChunkFloats / (kBlock * 4);  // 4 x b128 per thread

// CDNA5 async copy engine: GLOBAL_LOAD_ASYNC_TO_LDS_B128 (tracked with ASYNCcnt).
// vdst = per-lane LDS byte address, vaddr = per-lane 64-bit global address.
__device__ __forceinline__ void async_copy_b128(void* lds_dst, const void* gsrc) {
  // Low 32 bits of a flat LDS pointer are the LDS byte offset on gfx1250.
  uint32_t lds_off = (uint32_t)(uintptr_t)lds_dst;
  asm volatile("global_load_async_to_lds_b128 %0, %1, off"
               :
               : "v"(lds_off), "v"(gsrc)
               : "memory");
}

// Watermark wait: everything except the most recent 12 async copies has landed.
__device__ __forceinline__ void wait_async_le12() {
  asm volatile("s_wait_asynccnt 12" ::: "memory");
}
__device__ __forceinline__ void wait_async_0() {
  asm volatile("s_wait_asynccnt 0" ::: "memory");
}

__device__ __forceinline__ float fast_tanh(float v) {
#if __has_builtin(__builtin_amdgcn_tanhf)
  return __builtin_amdgcn_tanhf(v);  // v_tanh_f32 (TRANS pipe, co-executes w/ VALU)
#else
  // tanh(v) = 1 - 2 / (exp2(2*log2(e)*v) + 1)
  float e = __builtin_amdgcn_exp2f(v * 2.88539008177793f);
  return __builtin_fmaf(-2.0f, __builtin_amdgcn_rcpf(e + 1.0f), 1.0f);
#endif
}

}  // namespace

__global__ __launch_bounds__(kBlock) void kan_tanh_basis_kernel(
    const float* __restrict__ x,   // [R, I]
    const float* __restrict__ w,   // [I, N, O]
    const float* __restrict__ h,   // [I, N, O]
    const float* __restrict__ bb,  // [I, N, O]
    float* __restrict__ out) {     // [R, O]
  __shared__ __align__(16) float s_x[kTileRows * kI];   // 4 KB, staged once
  __shared__ __align__(16) float s_h[2][kChunkFloats];  // 32 KB, double buffered
  __shared__ __align__(16) float s_b[2][kChunkFloats];  // 32 KB
  __shared__ __align__(16) float s_w[2][kChunkFloats];  // 32 KB

  const int tid = threadIdx.x;
  const int o = tid & (kO - 1);   // lane -> output feature (contiguous)
  const int rq = tid >> 6;        // 0..3: row phase within the tile
  const int row0 = blockIdx.x * kTileRows;

  // Stage the x tile (16 rows x 64 = 1024 floats): one b128 per thread.
  async_copy_b128(&s_x[tid * 4], x + (size_t)row0 * kI + tid * 4);

  // Chunk stager: params are [I,N,O], so an i-chunk is 16 KB *contiguous* per array.
  auto stage_chunk = [&](int ci, int buf) {
#pragma unroll
    for (int j = 0; j < kCopiesPerArray; ++j) {
      const int off = (j * kBlock + tid) * 4;  // float offset, 16B aligned
      const size_t g = (size_t)ci * kChunkFloats + off;
      async_copy_b128(&s_h[buf][off], h + g);
      async_copy_b128(&s_b[buf][off], bb + g);
      async_copy_b128(&s_w[buf][off], w + g);
    }
  };

  stage_chunk(0, 0);  // 12 copies in flight (+1 for x)

  float acc[kRowsPerThread];
#pragma unroll
  for (int k = 0; k < kRowsPerThread; ++k) acc[k] = 0.0f;

  for (int ci = 0; ci < kNChunks; ++ci) {
    const int buf = ci & 1;
    if (ci + 1 < kNChunks) {
      // Prefetch next chunk into the other buffer (its previous readers passed
      // the trailing barrier of iteration ci-1), then retire this chunk's copies.
      stage_chunk(ci + 1, buf ^ 1);
      wait_async_le12();
    } else {
      wait_async_0();
    }
    __syncthreads();

#pragma unroll
    for (int ii = 0; ii < kCI; ++ii) {
      float xv[kRowsPerThread];
#pragma unroll
      for (int k = 0; k < kRowsPerThread; ++k)
        xv[k] = s_x[(rq + 4 * k) * kI + ci * kCI + ii];  // wave-uniform broadcast
#pragma unroll
      for (int n = 0; n < kN; ++n) {
        const int idx = (ii * kN + n) * kO + o;  // stride-1 across lanes
        const float hv = s_h[buf][idx];
        const float bv = s_b[buf][idx];
        const float wv = s_w[buf][idx];
#pragma unroll
        for (int k = 0; k < kRowsPerThread; ++k) {
          const float t = fast_tanh(__builtin_fmaf(xv[k], hv, bv));
          acc[k] = __builtin_fmaf(t, wv, acc[k]);
        }
      }
    }
    __syncthreads();  // protect buf before the next iteration's prefetch overwrites it
  }

#pragma unroll
  for (int k = 0; k < kRowsPerThread; ++k) {
    const int row = row0 + rq + 4 * k;
    out[(size_t)row * kO + o] = acc[k];
  }
}

extern "C" void kernel_launch(void* const* d_in, const int* in_sizes, int n_in,
                              void* d_out, int out_size, void* d_ws, size_t ws_size,
                              hipStream_t stream) {
  // setup_inputs() order: x, w, h, b (all float32)
  const float* x = (const float*)d_in[0];
  const float* w = (const float*)d_in[1];
  const float* h = (const float*)d_in[2];
  const float* b = (const float*)d_in[3];
  float* out = (float*)d_out;

  const int rows = in_sizes[0] / kI;   // B*S = 2048
  const int grid = rows / kTileRows;   // 128 blocks x 8 waves
  kan_tanh_basis_kernel<<<grid, kBlock, 0, stream>>>(x, w, h, b, out);
}